// GraphSAGEEmbedder_56160992363059
// MI455X (gfx1250) — compile-verified
//
#include <hip/hip_runtime.h>

typedef float v2f __attribute__((ext_vector_type(2)));
typedef float v8f __attribute__((ext_vector_type(8)));

#define DIM 128
#define AS_STRIDE 258   // 256 cols + pad 2: keeps b64 align, spreads LDS banks
#define ROWS_PER_BLOCK 64
#define ROW_TILES 4     // 4 x 16-row WMMA tiles per block

// ---------------- zero scratch ----------------
__global__ void zero_f32(float* __restrict__ p, long n) {
  long i = (long)blockIdx.x * blockDim.x + threadIdx.x;
  long stride = (long)gridDim.x * blockDim.x;
  for (; i < n; i += stride) p[i] = 0.0f;
}

// ---------------- edge scatter: sum[dst] += x[src]; cnt[dst] += 1 ----------------
// 32 lanes per edge, 4 columns per lane (float4 gather, 4x f32 atomics)
__global__ void scatter_mean(const float* __restrict__ x, const int* __restrict__ src,
                             const int* __restrict__ dst, float* __restrict__ sum,
                             float* __restrict__ cnt, int E) {
  int t = blockIdx.x * blockDim.x + threadIdx.x;
  int e = t >> 5;
  int lane = t & 31;
  if (e >= E) return;
  int s = src[e];
  int d = dst[e];
  const float4* xr = (const float4*)(x + (long)s * DIM);
  float4 v = xr[lane];
  float* sp = sum + (long)d * DIM + lane * 4;
  atomicAdd(sp + 0, v.x);
  atomicAdd(sp + 1, v.y);
  atomicAdd(sp + 2, v.z);
  atomicAdd(sp + 3, v.w);
  if (lane == 0) atomicAdd(cnt + d, 1.0f);
}

// ---------------- fused SAGE GEMM: out = relu?( (sum/cnt)@Wl + x@Wr + b ) ----------------
// block = 256 threads = 8 waves; block handles 64 rows (4 x 16-row WMMA tiles).
// Wave w owns output cols [16w,16w+16). Each B fragment (2 global loads) is reused
// across 4 independent accumulators -> 4x less W traffic + deep WMMA pipelining.
__global__ __launch_bounds__(256) void sage_gemm(
    const float* __restrict__ sum, const float* __restrict__ cnt,
    const float* __restrict__ xin, const float* __restrict__ Wl,
    const float* __restrict__ Wr, const float* __restrict__ bias,
    float* __restrict__ out, int n_tgt, int do_relu) {
  __shared__ float As[ROWS_PER_BLOCK * AS_STRIDE];   // 64 x 256 (+pad)  ~66 KB
  __shared__ float invc[ROWS_PER_BLOCK];

  int tid = threadIdx.x;
  int m0 = blockIdx.x * ROWS_PER_BLOCK;

  // Per-row 1/max(cnt,1)
  if (tid < ROWS_PER_BLOCK) {
    int row = m0 + tid;
    float cn = (row < n_tgt) ? cnt[row] : 1.0f;
    invc[tid] = 1.0f / fmaxf(cn, 1.0f);
  }
  __syncthreads();

  // Stage A tile: cols 0..127 = mean-aggregated neighbors, cols 128..255 = x (root)
  for (int i = tid; i < ROWS_PER_BLOCK * 256; i += 256) {
    int r = i >> 8;
    int c = i & 255;
    int row = m0 + r;
    float val = 0.0f;
    if (row < n_tgt) {
      if (c < DIM) {
        val = sum[(long)row * DIM + c] * invc[r];
      } else {
        val = xin[(long)row * DIM + (c - DIM)];
      }
    }
    As[r * AS_STRIDE + c] = val;
  }
  __syncthreads();

  int wave = tid >> 5;
  int lane = tid & 31;
  int nb = wave * 16;
  int mrow = lane & 15;            // A-frag row within 16-row tile
  int koff = (lane >> 4) << 1;     // lanes 0-15: K=0,1 ; lanes 16-31: K=2,3
  int ncol = nb + (lane & 15);     // this lane's output column

  v8f acc0 = {}, acc1 = {}, acc2 = {}, acc3 = {};
#pragma unroll
  for (int ph = 0; ph < 2; ++ph) {
    const float* __restrict__ W = (ph == 0) ? Wl : Wr;
    const float* __restrict__ abase = &As[mrow * AS_STRIDE + ph * DIM];
#pragma unroll 4
    for (int k = 0; k < DIM; k += 4) {
      v2f b;
      b.x = W[(long)(k + koff) * DIM + ncol];
      b.y = W[(long)(k + koff + 1) * DIM + ncol];
#define SAGE_TILE(T, ACC)                                                     \
      {                                                                       \
        float2 af = *(const float2*)(abase + (T) * 16 * AS_STRIDE + k + koff);\
        v2f a; a.x = af.x; a.y = af.y;                                        \
        ACC = __builtin_amdgcn_wmma_f32_16x16x4_f32(false, a, false, b,       \
                                                    (short)0, ACC, false, false); \
      }
      SAGE_TILE(0, acc0)
      SAGE_TILE(1, acc1)
      SAGE_TILE(2, acc2)
      SAGE_TILE(3, acc3)
#undef SAGE_TILE
    }
  }

  // C/D layout: lanes 0-15 -> N=lane, M=vgpr; lanes 16-31 -> N=lane-16, M=vgpr+8
  float bv = bias[ncol];
  int mbase = (lane >> 4) << 3;
  v8f accs[ROW_TILES] = {acc0, acc1, acc2, acc3};
#pragma unroll
  for (int t = 0; t < ROW_TILES; ++t) {
#pragma unroll
    for (int v = 0; v < 8; ++v) {
      int row = m0 + t * 16 + mbase + v;
      if (row < n_tgt) {
        float val = accs[t][v] + bv;
        if (do_relu) val = fmaxf(val, 0.0f);
        out[(long)row * DIM + ncol] = val;
      }
    }
  }
}

// ---------------- host side ----------------
extern "C" void kernel_launch(void* const* d_in, const int* in_sizes, int n_in,
                              void* d_out, int out_size, void* d_ws, size_t ws_size,
                              hipStream_t stream) {
  const float* x   = (const float*)d_in[0];
  const float* Wl1 = (const float*)d_in[1];
  const float* Wr1 = (const float*)d_in[2];
  const float* b1  = (const float*)d_in[3];
  const float* Wl2 = (const float*)d_in[4];
  const float* Wr2 = (const float*)d_in[5];
  const float* b2  = (const float*)d_in[6];
  const float* Wl3 = (const float*)d_in[7];
  const float* Wr3 = (const float*)d_in[8];
  const float* b3  = (const float*)d_in[9];
  const int* src1 = (const int*)d_in[10];
  const int* dst1 = (const int*)d_in[11];
  const int* src2 = (const int*)d_in[12];
  const int* dst2 = (const int*)d_in[13];
  const int* src3 = (const int*)d_in[14];
  const int* dst3 = (const int*)d_in[15];

  const int N1 = 80000, N2 = 16000, N3 = 4096;
  const int E1 = in_sizes[10], E2 = in_sizes[12], E3 = in_sizes[14];

  float* out = (float*)d_out;
  float* h1 = out;
  float* h2 = out + (long)N1 * DIM;
  float* h3 = out + (long)(N1 + N2) * DIM;

  float* sumB = (float*)d_ws;            // up to N1*128 floats (~41 MB)
  float* cntB = sumB + (long)N1 * DIM;   // up to N1 floats

  auto run_layer = [&](const float* xin, const int* src, const int* dst, int E,
                       int n_tgt, const float* Wl, const float* Wr, const float* b,
                       float* hout, int relu) {
    long sn = (long)n_tgt * DIM;
    zero_f32<<<2048, 256, 0, stream>>>(sumB, sn);
    zero_f32<<<256, 256, 0, stream>>>(cntB, (long)n_tgt);
    int sblocks = (E * 32 + 255) / 256;
    scatter_mean<<<sblocks, 256, 0, stream>>>(xin, src, dst, sumB, cntB, E);
    int gblocks = (n_tgt + ROWS_PER_BLOCK - 1) / ROWS_PER_BLOCK;
    sage_gemm<<<gblocks, 256, 0, stream>>>(sumB, cntB, xin, Wl, Wr, b, hout,
                                           n_tgt, relu);
  };

  run_layer(x,  src1, dst1, E1, N1, Wl1, Wr1, b1, h1, 1);
  run_layer(h1, src2, dst2, E2, N2, Wl2, Wr2, b2, h2, 1);
  run_layer(h2, src3, dst3, E3, N3, Wl3, Wr3, b3, h3, 0);
}